// GCN_54262616818367
// MI455X (gfx1250) — compile-verified
//
#include <hip/hip_runtime.h>
#include <hip/hip_bf16.h>

typedef float v2f __attribute__((ext_vector_type(2)));
typedef float v8f __attribute__((ext_vector_type(8)));

#define F_IN 512
#define HID  16
#define NCLS 7

// ---------------- degree / normalization ----------------

__global__ void k_deg_init(float* deg, int n) {
    int i = blockIdx.x * blockDim.x + threadIdx.x;
    if (i < n) deg[i] = 1.0f;               // self loop contributes 1
}

__global__ void k_deg_count(const int* __restrict__ col, float* __restrict__ deg, int E) {
    int e = blockIdx.x * blockDim.x + threadIdx.x;
    if (e < E) unsafeAtomicAdd(&deg[col[e]], 1.0f);
}

__global__ void k_rsqrt(float* deg, int n) {
    int i = blockIdx.x * blockDim.x + threadIdx.x;
    if (i < n) deg[i] = __frsqrt_rn(deg[i]);   // deg >= 1 always
}

// ---------------- layer 1 GEMM: Y = X @ W1 (Nx512 @ 512x16), fp32 WMMA ----------------
// One wave computes a 16(M) x 16(N) tile; K loop in steps of 4 via v_wmma_f32_16x16x4_f32.
// A 16x4 layout (ISA 7.12.2): lane m = lane&15; lanes<16 hold K={k,k+1}, lanes>=16 K={k+2,k+3}.
// B 4x16: lane n = lane&15, same K split. C/D: lane n = lane&15; vgpr v -> row v (+8 for hi lanes).

__global__ __launch_bounds__(256) void k_gemm1(const float* __restrict__ X,
                                               const float* __restrict__ W1,
                                               float* __restrict__ Y, int ntiles) {
    __shared__ float sW[F_IN * HID];                 // 32 KB of 320 KB LDS
    for (int i = threadIdx.x; i < F_IN * HID; i += 256) sW[i] = W1[i];
    __syncthreads();

    int wave = threadIdx.x >> 5;
    int tile = blockIdx.x * 8 + wave;
    if (tile >= ntiles) return;                      // wave-uniform: EXEC all-ones for WMMA

    int lane = threadIdx.x & 31;
    int m    = lane & 15;
    int kf   = (lane >> 4) << 1;                     // 0 or 2

    const float* xrow = X + (size_t)(tile * 16 + m) * F_IN + kf;
    v8f acc = {};
#pragma unroll 4
    for (int k = 0; k < F_IN; k += 4) {
        v2f a = *(const v2f*)(xrow + k);             // global_load_b64, walks a row
        v2f b;
        b.x = sW[(k + kf) * HID + m];                // conflict-free ds_load
        b.y = sW[(k + kf + 1) * HID + m];
        acc = __builtin_amdgcn_wmma_f32_16x16x4_f32(false, a, false, b,
                                                    (short)0, acc, false, false);
    }

    int rbase = tile * 16 + ((lane >> 4) << 3);
    float* yo = Y + (size_t)rbase * HID + m;
#pragma unroll
    for (int v = 0; v < 8; ++v) yo[v * HID] = acc[v];   // 16 lanes -> coalesced 64B rows
}

// ---------------- layer 1 epilogue: self loop + bias, edge scatter, relu ----------------

__global__ void k_selfbias1(const float* __restrict__ Y, const float* __restrict__ dinv,
                            const float* __restrict__ b1, float* __restrict__ H, int n16) {
    int tid = blockIdx.x * blockDim.x + threadIdx.x;
    if (tid >= n16) return;
    int i = tid >> 4, t = tid & 15;
    float di = dinv[i];
    H[tid] = Y[tid] * di * di + b1[t];
}

__global__ void k_edge16(const int* __restrict__ ei, const float* __restrict__ Y,
                         const float* __restrict__ dinv, float* __restrict__ H, int E) {
    int tid = blockIdx.x * blockDim.x + threadIdx.x;
    int e = tid >> 4;
    if (e >= E) return;
    int t = tid & 15;
    int r = ei[e], c = ei[E + e];
    float w = dinv[r] * dinv[c];
    unsafeAtomicAdd(&H[(size_t)c * HID + t], Y[(size_t)r * HID + t] * w);
}

__global__ void k_relu(float* H, int n16) {
    int tid = blockIdx.x * blockDim.x + threadIdx.x;
    if (tid < n16) H[tid] = fmaxf(H[tid], 0.0f);
}

// ---------------- layer 2 GEMM: Z0 = H @ W2 (Nx16 @ 16x7), padded to N=16 ----------------

__global__ __launch_bounds__(256) void k_gemm2(const float* __restrict__ H,
                                               const float* __restrict__ W2,
                                               float* __restrict__ Z0, int ntiles) {
    __shared__ float sW[16 * 16];                    // W2 zero-padded 16x16
    {
        int kk = threadIdx.x >> 4, nn = threadIdx.x & 15;
        sW[threadIdx.x] = (nn < NCLS) ? W2[kk * NCLS + nn] : 0.0f;
    }
    __syncthreads();

    int wave = threadIdx.x >> 5;
    int tile = blockIdx.x * 8 + wave;
    if (tile >= ntiles) return;

    int lane = threadIdx.x & 31;
    int m    = lane & 15;
    int kf   = (lane >> 4) << 1;

    const float* hrow = H + (size_t)(tile * 16 + m) * HID + kf;
    v8f acc = {};
#pragma unroll
    for (int k = 0; k < HID; k += 4) {
        v2f a = *(const v2f*)(hrow + k);
        v2f b;
        b.x = sW[(k + kf) * 16 + m];
        b.y = sW[(k + kf + 1) * 16 + m];
        acc = __builtin_amdgcn_wmma_f32_16x16x4_f32(false, a, false, b,
                                                    (short)0, acc, false, false);
    }

    if (m < NCLS) {                                  // divergence only after WMMA: legal
        int rbase = tile * 16 + ((lane >> 4) << 3);
        float* zo = Z0 + (size_t)rbase * NCLS + m;
#pragma unroll
        for (int v = 0; v < 8; ++v) zo[v * NCLS] = acc[v];
    }
}

// ---------------- layer 2 epilogue ----------------

__global__ void k_selfbias2(const float* __restrict__ Z0, const float* __restrict__ dinv,
                            const float* __restrict__ b2, float* __restrict__ Z, int n) {
    int tid = blockIdx.x * blockDim.x + threadIdx.x;
    int i = tid >> 3, t = tid & 7;
    if (i >= n || t >= NCLS) return;
    float di = dinv[i];
    Z[i * NCLS + t] = Z0[i * NCLS + t] * di * di + b2[t];
}

__global__ void k_edge7(const int* __restrict__ ei, const float* __restrict__ Z0,
                        const float* __restrict__ dinv, float* __restrict__ Z, int E) {
    int tid = blockIdx.x * blockDim.x + threadIdx.x;
    int e = tid >> 3, t = tid & 7;
    if (e >= E || t >= NCLS) return;
    int r = ei[e], c = ei[E + e];
    float w = dinv[r] * dinv[c];
    unsafeAtomicAdd(&Z[(size_t)c * NCLS + t], Z0[(size_t)r * NCLS + t] * w);
}

// ---------------- launcher ----------------

extern "C" void kernel_launch(void* const* d_in, const int* in_sizes, int n_in,
                              void* d_out, int out_size, void* d_ws, size_t ws_size,
                              hipStream_t stream) {
    const float* X  = (const float*)d_in[0];
    const float* W1 = (const float*)d_in[1];
    const float* b1 = (const float*)d_in[2];
    const float* W2 = (const float*)d_in[3];
    const float* b2 = (const float*)d_in[4];
    const int*   ei = (const int*)d_in[5];

    const int N = in_sizes[0] / F_IN;      // 100000
    const int E = in_sizes[5] / 2;         // 3200000
    const int ntiles = N / 16;             // 6250 (N multiple of 16)

    // workspace: dinv[N] | Y[N*16] | Z0[N*7]
    float* dinv = (float*)d_ws;
    float* Y    = dinv + N;
    float* Z0   = Y + (size_t)N * HID;

    float* Hout = (float*)d_out;                   // h: N*16 (post-relu)
    float* Zout = Hout + (size_t)N * HID;          // z: N*7

    const int TB = 256;

    // normalization coefficients
    k_deg_init <<<(N + TB - 1) / TB, TB, 0, stream>>>(dinv, N);
    k_deg_count<<<(E + TB - 1) / TB, TB, 0, stream>>>(ei + E, dinv, E);
    k_rsqrt    <<<(N + TB - 1) / TB, TB, 0, stream>>>(dinv, N);

    // layer 1
    k_gemm1    <<<(ntiles + 7) / 8, TB, 0, stream>>>(X, W1, Y, ntiles);
    k_selfbias1<<<(N * HID + TB - 1) / TB, TB, 0, stream>>>(Y, dinv, b1, Hout, N * HID);
    k_edge16   <<<((size_t)E * 16 + TB - 1) / TB, TB, 0, stream>>>(ei, Y, dinv, Hout, E);
    k_relu     <<<(N * HID + TB - 1) / TB, TB, 0, stream>>>(Hout, N * HID);

    // layer 2
    k_gemm2    <<<(ntiles + 7) / 8, TB, 0, stream>>>(Hout, W2, Z0, ntiles);
    k_selfbias2<<<(N * 8 + TB - 1) / TB, TB, 0, stream>>>(Z0, dinv, b2, Zout, N);
    k_edge7    <<<((size_t)E * 8 + TB - 1) / TB, TB, 0, stream>>>(ei, Z0, dinv, Zout, E);
}